// Sinkhorn_51969104281659
// MI455X (gfx1250) — compile-verified
//
#include <hip/hip_runtime.h>
#include <hip/hip_bf16.h>

// MI455X / gfx1250 implementation of the Sinkhorn reference.
// Phase 1: build Kxx/Kyx/Kxy/Kyy with f32 WMMA (V_WMMA_F32_16X16X4_F32),
//          folding |x|^2+|y|^2-2x.y -> -exp(-dist) + log-weight. NT stores.
// Phase 2: 10 fixed-point iterations of row-wise streaming logsumexp.
//          L2 residency plan (192 MB global L2): Gxx+Gyy (128 MB) read with
//          regular temporal hints -> stay L2-resident across iterations;
//          Gyx+Gxy streamed with non-temporal hints so they don't evict them.
// Phase 3: final LSE pass (alpha=0) + weighted reduction to 4 scalars.

typedef __attribute__((ext_vector_type(2))) float v2f;
typedef __attribute__((ext_vector_type(4))) float v4f;
typedef __attribute__((ext_vector_type(8))) float v8f;

#define LDIM 2048
#define DDIM 64
#define BATCH 4

// ---------------------------------------------------------------------------
// Squared row norms: sq[b*L + i] = sum_d X[b,i,d]^2
// ---------------------------------------------------------------------------
__global__ void sqnorm_kernel(const float* __restrict__ X, float* __restrict__ sq) {
  const int i = blockIdx.x * blockDim.x + threadIdx.x;  // 0 .. B*L-1
  const float* xr = X + (size_t)i * DDIM;
  float acc = 0.f;
#pragma unroll
  for (int d = 0; d < DDIM; d += 4) {
    v4f v = *(const v4f*)(xr + d);
    acc += v.x * v.x + v.y * v.y + v.z * v.z + v.w * v.w;
  }
  sq[i] = acc;
}

// ---------------------------------------------------------------------------
// Gram + Gaussian kernel matrix via f32 WMMA.
// One wave computes one 16x16 tile of G[b] = -exp(-(|p|^2+|q|^2-2 p.q)/2) + w[col].
// A fragment: 16x4 f32 (2 VGPRs, lane m=lane&15, k = 2*(lane>>4)+v), ISA 7.12.2.
// B fragment: 4x16 f32, mirrored layout (n = lane&15, k = 2*(lane>>4)+v).
// K = 64 -> 16 WMMA steps, each fed by one 8-byte load per lane per operand.
// ---------------------------------------------------------------------------
__global__ void gram_gauss_kernel(const float* __restrict__ P, const float* __restrict__ Q,
                                  const float* __restrict__ sqP, const float* __restrict__ sqQ,
                                  const float* __restrict__ wcol, float* __restrict__ G) {
  const int gw   = blockIdx.x * 8 + (threadIdx.x >> 5);  // global wave id
  const int lane = threadIdx.x & 31;
  const int b  = gw >> 14;            // / (128*128)
  const int t  = gw & 16383;
  const int tr = t >> 7;              // row tile
  const int tc = t & 127;             // col tile
  const int m  = lane & 15;
  const int g  = lane >> 4;

  const float* Pb = P + (size_t)b * LDIM * DDIM;
  const float* Qb = Q + (size_t)b * LDIM * DDIM;
  const float* pA = Pb + (size_t)(tr * 16 + m) * DDIM + 2 * g;
  const float* pB = Qb + (size_t)(tc * 16 + m) * DDIM + 2 * g;

  v8f c = {};
#pragma unroll
  for (int s = 0; s < 16; ++s) {
    v2f av = *(const v2f*)(pA + 4 * s);
    v2f bv = *(const v2f*)(pB + 4 * s);
    // D = A x B + C  (16x16x4, all f32)
    c = __builtin_amdgcn_wmma_f32_16x16x4_f32(false, av, false, bv,
                                              (short)0, c, false, false);
  }

  const float* sqPb = sqP + b * LDIM;
  const float* sqQb = sqQ + b * LDIM;
  const float* wb   = wcol + b * LDIM;
  float* Gb = G + (size_t)b * LDIM * LDIM;

  const int col  = tc * 16 + m;       // C/D layout: N = lane&15
  const float sqc = sqQb[col];
  const float wc  = wb[col];
#pragma unroll
  for (int v = 0; v < 8; ++v) {
    const int row = tr * 16 + v + 8 * g;   // C/D layout: M = v + 8*(lane>>4)
    // dist = (|p|^2 + |q|^2 - 2 p.q) / (2*sigma^2*eps), sigma=eps=1
    const float dist = 0.5f * (sqPb[row] + sqc) - c[v];
    // NT store: written once, 256MB total can't live in L2 until Phase 2 anyway.
    __builtin_nontemporal_store(-expf(-dist) + wc, &Gb[(size_t)row * LDIM + col]);
  }
}

// ---------------------------------------------------------------------------
// Wave32 butterfly reductions.
// ---------------------------------------------------------------------------
__device__ inline float wave_max(float v) {
#pragma unroll
  for (int off = 16; off > 0; off >>= 1) v = fmaxf(v, __shfl_xor(v, off, 32));
  return v;
}
__device__ inline float wave_sum(float v) {
#pragma unroll
  for (int off = 16; off > 0; off >>= 1) v += __shfl_xor(v, off, 32);
  return v;
}

// ---------------------------------------------------------------------------
// Row-wise logsumexp + damped update:
//   n         = -log( sum_k exp(G[row,k] + pot[b,k]) )
//   fout[row] = alpha*fold[row] + (1-alpha)*n
// One block (256 threads = 8 waves) per row. Each thread owns 8 consecutive
// columns via two 16B loads (one wave spans a contiguous 1KB row chunk); the
// 8 values stay in registers across the max and exp-sum phases, so each
// matrix element is read from memory exactly once per pass.
// NT=true  -> non-temporal G loads (streamed matrices Gyx/Gxy)
// NT=false -> regular loads (Gxx/Gyy stay resident in the 192MB L2)
// ---------------------------------------------------------------------------
template <bool NT>
__global__ void lse_update_kernel(const float* __restrict__ G, const float* __restrict__ pot,
                                  const float* __restrict__ fold, float* __restrict__ fout,
                                  float alpha) {
  __shared__ float smax[8];
  __shared__ float ssum[8];
  const int row  = blockIdx.x;                 // 0 .. B*L-1
  const int tid  = threadIdx.x;
  const int wave = tid >> 5;
  const int lane = tid & 31;

  const float* Gr   = G + (size_t)row * LDIM + tid * 8;
  const float* potb = pot + (row & ~(LDIM - 1)) + tid * 8;  // pot for batch row/2048

  v4f g0, g1;
  if (NT) {
    g0 = __builtin_nontemporal_load((const v4f*)Gr);
    g1 = __builtin_nontemporal_load((const v4f*)Gr + 1);
  } else {
    g0 = *((const v4f*)Gr);
    g1 = *((const v4f*)Gr + 1);
  }
  const v4f p0 = *((const v4f*)potb);
  const v4f p1 = *((const v4f*)potb + 1);

  float vals[8];
#pragma unroll
  for (int i = 0; i < 4; ++i) { vals[i] = g0[i] + p0[i]; vals[4 + i] = g1[i] + p1[i]; }

  float m = vals[0];
#pragma unroll
  for (int i = 1; i < 8; ++i) m = fmaxf(m, vals[i]);
  m = wave_max(m);
  if (lane == 0) smax[wave] = m;
  __syncthreads();
#pragma unroll
  for (int j = 0; j < 8; ++j) m = fmaxf(m, smax[j]);

  float s = 0.f;
#pragma unroll
  for (int i = 0; i < 8; ++i) s += expf(vals[i] - m);
  s = wave_sum(s);
  if (lane == 0) ssum[wave] = s;
  __syncthreads();

  if (tid == 0) {
    float tot = 0.f;
#pragma unroll
    for (int j = 0; j < 8; ++j) tot += ssum[j];
    const float n = -(m + logf(tot));
    fout[row] = alpha * fold[row] + (1.0f - alpha) * n;
  }
}

// ---------------------------------------------------------------------------
// out[b] = sum_l (nxy - nxx/2)*a + sum_k (nyx - nyy/2)*b    (EPSILON = 1)
// ---------------------------------------------------------------------------
__global__ void finalize_kernel(const float* __restrict__ nxx, const float* __restrict__ nyx,
                                const float* __restrict__ nxy, const float* __restrict__ nyy,
                                const float* __restrict__ aw, const float* __restrict__ bw,
                                float* __restrict__ out) {
  __shared__ float ssum[8];
  const int bidx = blockIdx.x;
  const int tid  = threadIdx.x;
  const int base = bidx * LDIM;
  float acc = 0.f;
  for (int i = tid; i < LDIM; i += 256) {
    const float f = nxy[base + i] - 0.5f * nxx[base + i];
    const float g = nyx[base + i] - 0.5f * nyy[base + i];
    acc += f * aw[base + i] + g * bw[base + i];
  }
  acc = wave_sum(acc);
  if ((tid & 31) == 0) ssum[tid >> 5] = acc;
  __syncthreads();
  if (tid == 0) {
    float tot = 0.f;
#pragma unroll
    for (int j = 0; j < 8; ++j) tot += ssum[j];
    out[bidx] = tot;
  }
}

// ---------------------------------------------------------------------------
extern "C" void kernel_launch(void* const* d_in, const int* in_sizes, int n_in,
                              void* d_out, int out_size, void* d_ws, size_t ws_size,
                              hipStream_t stream) {
  (void)in_sizes; (void)n_in; (void)out_size; (void)ws_size;
  const float* x  = (const float*)d_in[0];   // (B, L, D)
  const float* a  = (const float*)d_in[1];   // (B, L) log-weights
  const float* y  = (const float*)d_in[2];   // (B, K, D)
  const float* bw = (const float*)d_in[3];   // (B, K) log-weights
  float* out = (float*)d_out;                // (B,)

  float* W = (float*)d_ws;
  const size_t MSZ = (size_t)BATCH * LDIM * LDIM;  // per-matrix floats (16.78M)
  const size_t VEC = (size_t)BATCH * LDIM;         // per-vector floats (8192)

  float* Gxx = W;                    // L2-resident pair: Gxx + Gyy (128 MB)
  float* Gyx = W + 1 * MSZ;          // streamed (NT) pair: Gyx + Gxy
  float* Gxy = W + 2 * MSZ;
  float* Gyy = W + 3 * MSZ;
  float* sqx = W + 4 * MSZ;
  float* sqy = sqx + VEC;
  float* fb  = sqy + VEC;
  float* fxx[2] = {fb + 0 * VEC, fb + 1 * VEC};
  float* fyx[2] = {fb + 2 * VEC, fb + 3 * VEC};
  float* fxy[2] = {fb + 4 * VEC, fb + 5 * VEC};
  float* fyy[2] = {fb + 6 * VEC, fb + 7 * VEC};
  float* nxx = fb + 8 * VEC;
  float* nyx = fb + 9 * VEC;
  float* nxy = fb + 10 * VEC;
  float* nyy = fb + 11 * VEC;

  // Phase 1: squared norms + the four 4x2048x2048 kernel matrices (WMMA).
  sqnorm_kernel<<<32, 256, 0, stream>>>(x, sqx);
  sqnorm_kernel<<<32, 256, 0, stream>>>(y, sqy);
  const int gramBlocks = (BATCH * 128 * 128) / 8;  // 8 waves (tiles) per block
  gram_gauss_kernel<<<gramBlocks, 256, 0, stream>>>(x, x, sqx, sqx, a,  Gxx);
  gram_gauss_kernel<<<gramBlocks, 256, 0, stream>>>(y, x, sqy, sqx, a,  Gyx);
  gram_gauss_kernel<<<gramBlocks, 256, 0, stream>>>(x, y, sqx, sqy, bw, Gxy);
  gram_gauss_kernel<<<gramBlocks, 256, 0, stream>>>(y, y, sqy, sqy, bw, Gyy);

  // Potentials start at zero.
  hipMemsetAsync(fb, 0, 8 * VEC * sizeof(float), stream);

  // Phase 2: 10 damped Sinkhorn iterations (reference dict order xx,yx,xy,yy;
  // nxy consumes the freshly averaged fyx -> use the [1-p] buffer).
  int p = 0;
  for (int it = 0; it < 10; ++it) {
    lse_update_kernel<false><<<BATCH * LDIM, 256, 0, stream>>>(Gxx, fxx[p],     fxx[p], fxx[1 - p], 0.5f);
    lse_update_kernel<true ><<<BATCH * LDIM, 256, 0, stream>>>(Gyx, fxy[p],     fyx[p], fyx[1 - p], 0.5f);
    lse_update_kernel<true ><<<BATCH * LDIM, 256, 0, stream>>>(Gxy, fyx[1 - p], fxy[p], fxy[1 - p], 0.5f);
    lse_update_kernel<false><<<BATCH * LDIM, 256, 0, stream>>>(Gyy, fyy[p],     fyy[p], fyy[1 - p], 0.5f);
    p ^= 1;
  }

  // Phase 3: final (gradient-carrying) extrapolation, alpha = 0.
  lse_update_kernel<false><<<BATCH * LDIM, 256, 0, stream>>>(Gxx, fxx[p], fxx[p], nxx, 0.0f);
  lse_update_kernel<true ><<<BATCH * LDIM, 256, 0, stream>>>(Gyx, fxy[p], fxy[p], nyx, 0.0f);
  lse_update_kernel<true ><<<BATCH * LDIM, 256, 0, stream>>>(Gxy, nyx,    nyx,    nxy, 0.0f);
  lse_update_kernel<false><<<BATCH * LDIM, 256, 0, stream>>>(Gyy, fyy[p], fyy[p], nyy, 0.0f);

  finalize_kernel<<<BATCH, 256, 0, stream>>>(nxx, nyx, nxy, nyy, a, bw, out);
}